// ParcorCoefficientsToLinearPredictiveCoefficients_55009941127436
// MI455X (gfx1250) — compile-verified
//
#include <hip/hip_runtime.h>
#include <cstdint>
#include <cstddef>

// ParcorCoefficientsToLinearPredictiveCoefficients (Levinson step-up recursion)
// rows = 16*65536, each row has 33 fp32 coefficients.
// Memory-bound (~277 MB total traffic -> ~12 us floor @ 23.3 TB/s).
// Strategy: stage 256 rows per block through LDS using CDNA5 async
// global<->LDS DMA (coalesced b128), do the 496-FMA recursion per row in VGPRs.

#define ORDER1 33                      // LPC_ORDER + 1
#define ROWS_PER_BLOCK 256
#define BLOCK_DWORDS (ROWS_PER_BLOCK * ORDER1)   // 8448 dwords = 33792 bytes

__device__ __forceinline__ void levinson_rec(float a[ORDER1]) {
    // a[1:m] += k[m] * reverse(a[1:m]) for m = 2..32, using pre-update values.
#pragma unroll
    for (int m = 2; m < ORDER1; ++m) {
        const float km = a[m];   // a[m] still equals k[m] at step m
#pragma unroll
        for (int i = 1; 2 * i < m; ++i) {
            const float lo = a[i];
            const float hi = a[m - i];
            a[i]     = fmaf(km, hi, lo);
            a[m - i] = fmaf(km, lo, hi);
        }
        if ((m & 1) == 0) {
            const int mid = m >> 1;
            a[mid] = fmaf(km, a[mid], a[mid]);
        }
    }
}

__global__ __launch_bounds__(ROWS_PER_BLOCK)
void parcor_to_lpc_async_kernel(const float* __restrict__ src,
                                float* __restrict__ dst) {
    __shared__ float lds[BLOCK_DWORDS];

    const int tid = threadIdx.x;
    const uint64_t blockBase = (uint64_t)blockIdx.x * (uint64_t)BLOCK_DWORDS;
    const float* gsrc = src + blockBase;     // block-uniform -> SGPR pair
    float*       gdst = dst + blockBase;     // block-uniform -> SGPR pair

    // Low 32 bits of the generic pointer to a __shared__ object = LDS byte
    // offset relative to the wave's LDS base, which is what the async
    // instructions' LDS-address VGPR expects.
    const uint32_t lds0 = (uint32_t)(uintptr_t)(&lds[0]);

    // ---- Phase 1: coalesced async DMA global -> LDS (bypasses VGPRs) ----
    // 8 rounds of b128 (8192 dwords) + 1 round of b32 (256 dwords) = 8448.
#pragma unroll
    for (int it = 0; it < 8; ++it) {
        const uint32_t voff = (uint32_t)(it * 1024 + tid * 4) * 4u;  // byte offset
        const uint32_t la   = lds0 + voff;
        asm volatile("global_load_async_to_lds_b128 %0, %1, %2"
                     :: "v"(la), "v"(voff), "s"(gsrc) : "memory");
    }
    {
        const uint32_t voff = (uint32_t)(8192 + tid) * 4u;
        const uint32_t la   = lds0 + voff;
        asm volatile("global_load_async_to_lds_b32 %0, %1, %2"
                     :: "v"(la), "v"(voff), "s"(gsrc) : "memory");
    }
    asm volatile("s_wait_asynccnt 0" ::: "memory");
    __syncthreads();

    // ---- Phase 2: per-lane recursion entirely in VGPRs ----
    // Row t occupies lds[t*33 .. t*33+32]; (33*lane mod 64) is conflict-free.
    float a[ORDER1];
    const int rb = tid * ORDER1;
#pragma unroll
    for (int j = 0; j < ORDER1; ++j) a[j] = lds[rb + j];

    levinson_rec(a);

#pragma unroll
    for (int j = 0; j < ORDER1; ++j) lds[rb + j] = a[j];
    __syncthreads();   // emits s_wait_dscnt before barrier: LDS results visible

    // ---- Phase 3: coalesced async DMA LDS -> global ----
#pragma unroll
    for (int it = 0; it < 8; ++it) {
        const uint32_t voff = (uint32_t)(it * 1024 + tid * 4) * 4u;
        const uint32_t la   = lds0 + voff;
        asm volatile("global_store_async_from_lds_b128 %0, %1, %2"
                     :: "v"(voff), "v"(la), "s"(gdst) : "memory");
    }
    {
        const uint32_t voff = (uint32_t)(8192 + tid) * 4u;
        const uint32_t la   = lds0 + voff;
        asm volatile("global_store_async_from_lds_b32 %0, %1, %2"
                     :: "v"(voff), "v"(la), "s"(gdst) : "memory");
    }
    asm volatile("s_wait_asynccnt 0" ::: "memory");
}

// Guarded scalar fallback for any remainder rows (not hit for the reference
// shape, where rows % 256 == 0).
__global__ __launch_bounds__(256)
void parcor_to_lpc_tail_kernel(const float* __restrict__ src,
                               float* __restrict__ dst,
                               int row0, int nrows) {
    const int r = row0 + blockIdx.x * blockDim.x + threadIdx.x;
    if (r >= nrows) return;
    const float* p = src + (size_t)r * ORDER1;
    float*       q = dst + (size_t)r * ORDER1;
    float a[ORDER1];
#pragma unroll
    for (int j = 0; j < ORDER1; ++j) a[j] = p[j];
    levinson_rec(a);
#pragma unroll
    for (int j = 0; j < ORDER1; ++j) q[j] = a[j];
}

extern "C" void kernel_launch(void* const* d_in, const int* in_sizes, int n_in,
                              void* d_out, int out_size, void* d_ws, size_t ws_size,
                              hipStream_t stream) {
    (void)n_in; (void)out_size; (void)d_ws; (void)ws_size;
    const float* k   = (const float*)d_in[0];
    float*       out = (float*)d_out;

    const int rows    = in_sizes[0] / ORDER1;          // 1,048,576 for reference
    const int nblocks = rows / ROWS_PER_BLOCK;         // 4096 full blocks

    if (nblocks > 0) {
        parcor_to_lpc_async_kernel<<<nblocks, ROWS_PER_BLOCK, 0, stream>>>(k, out);
    }
    const int done = nblocks * ROWS_PER_BLOCK;
    const int rem  = rows - done;
    if (rem > 0) {
        const int tb = (rem + 255) / 256;
        parcor_to_lpc_tail_kernel<<<tb, 256, 0, stream>>>(k, out, done, rows);
    }
}